// ValueLayerSlow_11063835754519
// MI455X (gfx1250) — compile-verified
//
#include <hip/hip_runtime.h>
#include <hip/hip_bf16.h>
#include <math.h>

typedef __attribute__((ext_vector_type(2))) float v2f;
typedef __attribute__((ext_vector_type(8))) float v8f;

#define TOTAL 10
#define DIM 1024           // 2^10
#define ENC 32
#define NPAIR 4096         // B*T*T = 4*32*32
#define C_IN 512
#define N_OUT 2048         // 2*DIM (re/im interleaved)

// ---------------------------------------------------------------------------
// K1: enc[row, :] = normalize(x[row, :] @ W^T + b),  row = b*32 + t  (128 rows)
// ---------------------------------------------------------------------------
__global__ void encode_kernel(const float* __restrict__ x,
                              const float* __restrict__ W,
                              const float* __restrict__ bias,
                              float* __restrict__ enc) {
    __shared__ float sEnc[ENC];
    const int row = blockIdx.x;                 // 0..127
    const int t = threadIdx.x;
    const int o = t >> 3;                       // output channel 0..31
    const int k = t & 7;                        // sub-lane 0..7
    const float* xr = x + (size_t)row * C_IN;
    const float* wr = W + (size_t)o * C_IN;
    float acc = 0.f;
    for (int c = k; c < C_IN; c += 8)
        acc += xr[c] * wr[c];
    acc += __shfl_xor(acc, 1);
    acc += __shfl_xor(acc, 2);
    acc += __shfl_xor(acc, 4);
    if (k == 0) sEnc[o] = acc + bias[o];
    __syncthreads();
    if (t < ENC) {
        float v = sEnc[t];
        float ss = v * v;
        ss += __shfl_xor(ss, 1);
        ss += __shfl_xor(ss, 2);
        ss += __shfl_xor(ss, 4);
        ss += __shfl_xor(ss, 8);
        ss += __shfl_xor(ss, 16);
        enc[(size_t)row * ENC + t] = v * (1.0f / sqrtf(ss));
    }
}

// ---------------------------------------------------------------------------
// K2: build the 1024x1024 circuit operator U by simulating the circuit on
// basis vectors. Block y computes column y of U in LDS (8KB) and stores it as
// row y of B with interleaved re/im:
//    B[y][2x+0] = Re U[x][y],  B[y][2x+1] = Im U[x][y]
// so the downstream real GEMM writes complex64 output directly.
// Wire j lives on bit (9-j) (wire 0 = MSB, matching the reference reshape).
// ---------------------------------------------------------------------------
__global__ void build_u_kernel(const float* __restrict__ rx0,
                               const float* __restrict__ ry0,
                               const float* __restrict__ ry1,
                               float* __restrict__ Bmat) {
    __shared__ float2 st[DIM];
    const int t = threadIdx.x;                  // 0..511
    const int y = blockIdx.x;                   // 0..1023
    st[t]       = make_float2((t == y)       ? 1.f : 0.f, 0.f);
    st[t + 512] = make_float2((t + 512 == y) ? 1.f : 0.f, 0.f);
    __syncthreads();

    // ---- layer 1: RX(rx0[j]) then RY(ry0[j]) on each wire ----
    for (int j = 0; j < TOTAL; ++j) {
        const int p = 9 - j;
        const int lowmask = (1 << p) - 1;
        const int i0 = ((t & ~lowmask) << 1) | (t & lowmask);
        const int i1 = i0 | (1 << p);
        {   // RX: [[c, -is], [-is, c]]
            const float th = rx0[j] * 0.5f;
            const float cc = cosf(th), ssn = sinf(th);
            float2 s0 = st[i0], s1 = st[i1];
            float2 n0 = make_float2(cc * s0.x + ssn * s1.y, cc * s0.y - ssn * s1.x);
            float2 n1 = make_float2(ssn * s0.y + cc * s1.x, -ssn * s0.x + cc * s1.y);
            st[i0] = n0; st[i1] = n1;
        }
        __syncthreads();
        {   // RY: [[c, -s], [s, c]]
            const float th = ry0[j] * 0.5f;
            const float cc = cosf(th), ssn = sinf(th);
            float2 s0 = st[i0], s1 = st[i1];
            float2 n0 = make_float2(cc * s0.x - ssn * s1.x, cc * s0.y - ssn * s1.y);
            float2 n1 = make_float2(ssn * s0.x + cc * s1.x, ssn * s0.y + cc * s1.y);
            st[i0] = n0; st[i1] = n1;
        }
        __syncthreads();
    }

    // ---- CNOT ring: control j -> target (j+1)%10, sequential ----
    for (int j = 0; j < TOTAL; ++j) {
        const int cbit = 9 - j;
        const int tbit = 9 - ((j + 1) % TOTAL);
        const int lowmask = (1 << tbit) - 1;
        const int i0 = ((t & ~lowmask) << 1) | (t & lowmask);
        const int i1 = i0 | (1 << tbit);
        if ((i0 >> cbit) & 1) {
            float2 tmp = st[i0]; st[i0] = st[i1]; st[i1] = tmp;
        }
        __syncthreads();
    }

    // ---- layer 2: RY(ry1[j]) on each wire ----
    for (int j = 0; j < TOTAL; ++j) {
        const int p = 9 - j;
        const int lowmask = (1 << p) - 1;
        const int i0 = ((t & ~lowmask) << 1) | (t & lowmask);
        const int i1 = i0 | (1 << p);
        const float th = ry1[j] * 0.5f;
        const float cc = cosf(th), ssn = sinf(th);
        float2 s0 = st[i0], s1 = st[i1];
        float2 n0 = make_float2(cc * s0.x - ssn * s1.x, cc * s0.y - ssn * s1.y);
        float2 n1 = make_float2(ssn * s0.x + cc * s1.x, ssn * s0.y + cc * s1.y);
        st[i0] = n0; st[i1] = n1;
        __syncthreads();
    }

    float2* Bf2 = (float2*)Bmat;                // row y, interleaved re/im
    Bf2[(size_t)y * DIM + t]       = st[t];
    Bf2[(size_t)y * DIM + t + 512] = st[t + 512];
}

// ---------------------------------------------------------------------------
// K3: Out(4096 x 2048) = S0(4096 x 1024) x B(1024 x 2048), f32 WMMA 16x16x4.
// S0[s][y] = enc[s>>5][y>>5] * enc[(s>>10)*32 + (s&31)][y&31], built on the
// fly from LDS-staged enc rows.
//
// M-blocking: each wave computes a 64x16 output slab (4 stacked 16x16 tiles,
// 4 independent accumulator chains) so each B operand pair is reused 4x ->
// 4x less B/L2 traffic, compute-bound loop: 2 loads + 4 pk_mul + 4 WMMA.
// Rows r and r+32 of the slab share the same enc_j row (s&31 has period 32),
// so only 32 ej rows + 2 ei rows are staged per wave.
//
// A layout: lane m=l&15, VGPR0/1 hold K = 2*(l>>4) and +1.
// B layout: lane n=l&15, VGPR0/1 hold K = 2*(l>>4) and +1.
// C/D layout: VGPR v -> row M = v + 8*(l>>4), col N = l&15.
// ---------------------------------------------------------------------------
__global__ void gemm_kernel(const float* __restrict__ enc,
                            const float* __restrict__ Bmat,
                            float* __restrict__ out) {
    __shared__ float ej_s[4][32][34];           // stride 34: conflict-free + 8B aligned pairs
    __shared__ float ei_s[4][2][32];
    const int lane = threadIdx.x & 31;
    const int w = threadIdx.x >> 5;             // wave id 0..3
    const int m0 = blockIdx.y * 256 + w * 64;   // output row base (pair index)
    const int n0 = blockIdx.x * 16;             // output col base (re/im cols)

    // stage the 32 distinct enc_j rows and the 2 enc_i rows for this slab
    for (int idx = lane; idx < 32 * 32; idx += 32) {
        const int r = idx >> 5, c = idx & 31;
        const int s = m0 + r;
        const int jrow = (s >> 10) * 32 + (s & 31);
        ej_s[w][r][c] = enc[(size_t)jrow * 32 + c];
    }
    ei_s[w][0][lane] = enc[(size_t)(m0 >> 5) * 32 + lane];
    ei_s[w][1][lane] = enc[(size_t)((m0 >> 5) + 1) * 32 + lane];
    __syncthreads();

    const int m = lane & 15;
    const int kh2 = (lane >> 4) * 2;            // 0 or 2
    const float* ej0 = &ej_s[w][m][0];          // rows of tiles 0 and 2
    const float* ej1 = &ej_s[w][16 + m][0];     // rows of tiles 1 and 3
    const float* ei0 = &ei_s[w][0][0];          // tiles 0,1
    const float* ei1 = &ei_s[w][1][0];          // tiles 2,3
    const float* bcol = Bmat + n0 + (lane & 15);

    v8f acc0 = {}, acc1 = {}, acc2 = {}, acc3 = {};
    #pragma unroll 4
    for (int k4 = 0; k4 < DIM; k4 += 4) {
        const int y0 = k4 + kh2;
        const float* bp = bcol + (size_t)y0 * N_OUT;
        v2f b; b.x = bp[0]; b.y = bp[N_OUT];

        const int c = y0 & 31;                  // even -> c+1 stays in row
        const float e0 = ei0[y0 >> 5];
        const float e1 = ei1[y0 >> 5];
        v2f j0; j0.x = ej0[c]; j0.y = ej0[c + 1];
        v2f j1; j1.x = ej1[c]; j1.y = ej1[c + 1];

        v2f a0; a0.x = e0 * j0.x; a0.y = e0 * j0.y;
        v2f a1; a1.x = e0 * j1.x; a1.y = e0 * j1.y;
        v2f a2; a2.x = e1 * j0.x; a2.y = e1 * j0.y;
        v2f a3; a3.x = e1 * j1.x; a3.y = e1 * j1.y;

        acc0 = __builtin_amdgcn_wmma_f32_16x16x4_f32(false, a0, false, b, (short)0, acc0, false, false);
        acc1 = __builtin_amdgcn_wmma_f32_16x16x4_f32(false, a1, false, b, (short)0, acc1, false, false);
        acc2 = __builtin_amdgcn_wmma_f32_16x16x4_f32(false, a2, false, b, (short)0, acc2, false, false);
        acc3 = __builtin_amdgcn_wmma_f32_16x16x4_f32(false, a3, false, b, (short)0, acc3, false, false);
    }

    const int ncol = n0 + (lane & 15);
    const int rhalf = (lane >> 4) * 8;
    float* o0 = out + (size_t)(m0 +   0 + rhalf) * N_OUT + ncol;
    float* o1 = out + (size_t)(m0 +  16 + rhalf) * N_OUT + ncol;
    float* o2 = out + (size_t)(m0 +  32 + rhalf) * N_OUT + ncol;
    float* o3 = out + (size_t)(m0 +  48 + rhalf) * N_OUT + ncol;
    #pragma unroll
    for (int v = 0; v < 8; ++v) {
        o0[(size_t)v * N_OUT] = acc0[v];
        o1[(size_t)v * N_OUT] = acc1[v];
        o2[(size_t)v * N_OUT] = acc2[v];
        o3[(size_t)v * N_OUT] = acc3[v];
    }
}

// ---------------------------------------------------------------------------
extern "C" void kernel_launch(void* const* d_in, const int* in_sizes, int n_in,
                              void* d_out, int out_size, void* d_ws, size_t ws_size,
                              hipStream_t stream) {
    const float* x     = (const float*)d_in[0];   // (4,32,512)
    const float* W_tok = (const float*)d_in[1];   // (32,512)
    const float* b_tok = (const float*)d_in[2];   // (32,)
    const float* rx0   = (const float*)d_in[3];   // (10,)
    const float* ry0   = (const float*)d_in[4];   // (10,)
    const float* ry1   = (const float*)d_in[5];   // (10,)
    float* out = (float*)d_out;                   // (4096, 2048) = complex64 (B,1024,1024)

    float* enc  = (float*)d_ws;                               // 128*32 f32 = 16 KB
    float* Bmat = (float*)((char*)d_ws + 16384);              // 1024*2048 f32 = 8 MB

    encode_kernel<<<128, 256, 0, stream>>>(x, W_tok, b_tok, enc);
    build_u_kernel<<<DIM, 512, 0, stream>>>(rx0, ry0, ry1, Bmat);
    gemm_kernel<<<dim3(N_OUT / 16, NPAIR / 256), 128, 0, stream>>>(enc, Bmat, out);
}